// Sparsify2D_abs_987842478202
// MI455X (gfx1250) — compile-verified
//
#include <hip/hip_runtime.h>
#include <stdint.h>

// Problem constants (match reference: x [32,256,56,56] fp32, SPARSE_RATIO=0.5)
#define HW       3136            // 56*56 elements per (b,c) row
#define CHUNKS   (HW / 4)        // 784 float4 chunks per row
#define K_RANK   1568            // int(0.5 * 3136): rank of threshold (k-th largest |x|)
#define NT       256             // threads per block (8 wave32s)
#define NW       8               // waves per block
#define HSTRIDE  264             // per-wave histogram stride (256 bins + 8 pad -> bank-shifted copies)

__global__ __launch_bounds__(NT) void sparsify_topk_kernel(const float* __restrict__ x,
                                                           float* __restrict__ out) {
  __shared__ float4   xs[CHUNKS];          // 12544 B: the staged row
  __shared__ unsigned hist[NW * HSTRIDE];  // per-wave 256-bin histograms (padded)
  __shared__ unsigned totals[NT];          // cross-wave reduced bins
  __shared__ unsigned suffix[NT + 1];      // suffix sums, suffix[256] = 0 sentinel
  __shared__ unsigned s_prefix, s_mask, s_kk;

  const int tid  = threadIdx.x;
  const int wave = tid >> 5;
  const int lane = tid & 31;
  const size_t rowbase = (size_t)blockIdx.x * (size_t)HW;

  // ---- Stage row into LDS with CDNA5 async global->LDS b128 copies (ASYNCcnt) ----
  {
    unsigned           lds0 = (unsigned)(uintptr_t)(&xs[0]);            // LDS byte offset
    unsigned long long g0   = (unsigned long long)(uintptr_t)(x + rowbase);
    for (int c = tid; c < CHUNKS; c += NT) {
      unsigned boff  = (unsigned)c * 16u;
      unsigned laddr = lds0 + boff;
      asm volatile("global_load_async_to_lds_b128 %0, %1, %2 offset:0"
                   :
                   : "v"(laddr), "v"(boff), "s"(g0)
                   : "memory");
    }
    asm volatile("s_wait_asynccnt 0x0" ::: "memory");  // this wave's async copies done
  }
  __syncthreads();  // all waves waited their ASYNCcnt -> full row visible in LDS

  // ---- 4-pass MSB-first radix select on abs-bits (uint order == float order for |x|) ----
  unsigned prefix = 0u, msk = 0u, kk = K_RANK;

  for (int shift = 24; shift >= 0; shift -= 8) {
    for (int i = tid; i < NW * HSTRIDE; i += NT) hist[i] = 0u;
    __syncthreads();

    unsigned* myhist = &hist[wave * HSTRIDE];
    for (int c = tid; c < CHUNKS; c += NT) {
      float4 v = xs[c];
      unsigned a;
      a = __float_as_uint(v.x) & 0x7FFFFFFFu;
      if ((a & msk) == prefix) atomicAdd(&myhist[(a >> shift) & 0xFFu], 1u);
      a = __float_as_uint(v.y) & 0x7FFFFFFFu;
      if ((a & msk) == prefix) atomicAdd(&myhist[(a >> shift) & 0xFFu], 1u);
      a = __float_as_uint(v.z) & 0x7FFFFFFFu;
      if ((a & msk) == prefix) atomicAdd(&myhist[(a >> shift) & 0xFFu], 1u);
      a = __float_as_uint(v.w) & 0x7FFFFFFFu;
      if ((a & msk) == prefix) atomicAdd(&myhist[(a >> shift) & 0xFFu], 1u);
    }
    __syncthreads();

    // Reduce the 8 wave-histograms: thread t owns bin t.
    unsigned tsum = 0u;
#pragma unroll
    for (int w = 0; w < NW; ++w) tsum += hist[w * HSTRIDE + tid];
    totals[tid] = tsum;
    __syncthreads();

    // Wave 0 computes suffix sums over the 256 bins (suffix[d] = count of bins >= d).
    if (wave == 0) {
      unsigned c[8], ls[8];
#pragma unroll
      for (int j = 0; j < 8; ++j) c[j] = totals[lane * 8 + j];
      ls[7] = c[7];
#pragma unroll
      for (int j = 6; j >= 0; --j) ls[j] = c[j] + ls[j + 1];
      unsigned tot = ls[0];          // this lane's 8-bin total
      unsigned inc = tot;            // inclusive suffix sum over lanes >= lane
#pragma unroll
      for (int d = 1; d < 32; d <<= 1) {
        unsigned o = __shfl_down(inc, (unsigned)d, 32);
        if (lane + d < 32) inc += o;
      }
      unsigned excl = inc - tot;     // sum over lanes strictly above
#pragma unroll
      for (int j = 0; j < 8; ++j) suffix[lane * 8 + j] = ls[j] + excl;
      if (lane == 0) suffix[NT] = 0u;
    }
    __syncthreads();

    // Unique selecting bin d: suffix[d] >= kk and suffix[d+1] < kk.
    unsigned sd  = suffix[tid];
    unsigned sd1 = suffix[tid + 1];
    if (sd >= kk && sd1 < kk) {
      s_prefix = prefix | ((unsigned)tid << shift);
      s_mask   = msk    | (0xFFu << shift);
      s_kk     = kk - sd1;   // rank within the selected bin's candidates
    }
    __syncthreads();
    prefix = (unsigned)__builtin_amdgcn_readfirstlane((int)s_prefix);
    msk    = (unsigned)__builtin_amdgcn_readfirstlane((int)s_mask);
    kk     = (unsigned)__builtin_amdgcn_readfirstlane((int)s_kk);
  }

  // prefix now holds the exact bit pattern of the k-th largest |x| in this row.
  const unsigned thresh = prefix;
  float* dst = out + rowbase;
  for (int c = tid; c < CHUNKS; c += NT) {
    float4 v = xs[c];
    v.x = ((__float_as_uint(v.x) & 0x7FFFFFFFu) >= thresh) ? v.x : 0.0f;
    v.y = ((__float_as_uint(v.y) & 0x7FFFFFFFu) >= thresh) ? v.y : 0.0f;
    v.z = ((__float_as_uint(v.z) & 0x7FFFFFFFu) >= thresh) ? v.z : 0.0f;
    v.w = ((__float_as_uint(v.w) & 0x7FFFFFFFu) >= thresh) ? v.w : 0.0f;
    ((float4*)dst)[c] = v;
  }
}

extern "C" void kernel_launch(void* const* d_in, const int* in_sizes, int n_in,
                              void* d_out, int out_size, void* d_ws, size_t ws_size,
                              hipStream_t stream) {
  (void)n_in; (void)out_size; (void)d_ws; (void)ws_size;
  const float* x = (const float*)d_in[0];
  float* out     = (float*)d_out;
  const int rows = in_sizes[0] / HW;   // 32*256 = 8192 (b,c) rows
  sparsify_topk_kernel<<<dim3(rows), dim3(NT), 0, stream>>>(x, out);
}